// MultiChainCondenseMSA_70085276336369
// MI455X (gfx1250) — compile-verified
//
#include <hip/hip_runtime.h>
#include <hip/hip_bf16.h>

// Problem constants (match the reference).
#define B_  16
#define NT_ 1024
#define TL_ 32
#define H_  64
#define S_  4096
// rows = B*NT*TL = 524288, each row = 64 f32 = 256 bytes.

// ---------------------------------------------------------------------------
// Kernel A: vectorized zero-fill (used for d_out and the count buffer).
// ---------------------------------------------------------------------------
__global__ __launch_bounds__(256) void zero4_kernel(uint4* __restrict__ p, unsigned n4) {
    unsigned t = blockIdx.x * 256u + threadIdx.x;
    if (t < n4) p[t] = make_uint4(0u, 0u, 0u, 0u);
}

// ---------------------------------------------------------------------------
// Kernel B: scatter-add with a per-wave double-buffered async global->LDS
// pipeline (gfx1250 GLOBAL_LOAD_ASYNC_TO_LDS_B128 + S_WAIT_ASYNCCNT).
//
// Each wave owns 8 consecutive rows, processed as 4 stages of 2 rows (512 B).
// Stage tile layout: lane L copies bytes [L*16, L*16+16) of the 512 B tile,
// so a wave consumes exactly the LDS bytes it loaded -> no cross-wave sync.
// Processing: lane L handles row (L>>4) of the tile, float4 group (L&15).
//
// The 128 MiB embedding stream is read-once -> th:TH_LOAD_NT so it does not
// evict the atomic accumulators (16 MiB out + 256 KiB counts) from the
// 192 MB L2, where all 33.5M global_atomic_add_f32 ops resolve.
// ---------------------------------------------------------------------------
__global__ __launch_bounds__(256) void scatter_kernel(
    const float* __restrict__ emb,   // [B, NT, TL, H] flattened, 256 B per row
    const int*   __restrict__ foc,   // [B, NT, TL] focus indices
    float*       __restrict__ out,   // [B, S, H]
    int*         __restrict__ cnt)   // [B, S]
{
    __shared__ float smem[8 * 256];          // 8 waves * 1024 B (2 x 512 B buffers)

    const int      lane  = threadIdx.x & 31;
    const int      wib   = threadIdx.x >> 5;             // wave in block (8 waves)
    const unsigned gwave = blockIdx.x * 8u + wib;
    const unsigned r0    = gwave * 8u;                   // first of this wave's 8 rows

    float*   wbase    = &smem[wib * 256];                // this wave's 1 KiB slice
    unsigned lds_base = (unsigned)(uintptr_t)wbase;      // LDS byte offset (addrspace(3) low bits)

    // Issue stage 0 into buffer 0.
    {
        const char* g = (const char*)emb + (size_t)r0 * 256u + (unsigned)lane * 16u;
        unsigned    l = lds_base + (unsigned)lane * 16u;
        asm volatile("global_load_async_to_lds_b128 %0, %1, off th:TH_LOAD_NT"
                     :: "v"(l), "v"(g) : "memory");
    }

    #pragma unroll
    for (int s = 0; s < 4; ++s) {
        if (s < 3) {
            // Prefetch next 2-row tile into the other buffer, then wait for the
            // current one (in-order async completion => asynccnt<=1 is enough).
            const char* g = (const char*)emb +
                            (size_t)(r0 + 2u * (unsigned)(s + 1)) * 256u + (unsigned)lane * 16u;
            unsigned    l = lds_base + (unsigned)((s + 1) & 1) * 512u + (unsigned)lane * 16u;
            asm volatile("global_load_async_to_lds_b128 %0, %1, off th:TH_LOAD_NT"
                         :: "v"(l), "v"(g) : "memory");
            asm volatile("s_wait_asynccnt 1" ::: "memory");
        } else {
            asm volatile("s_wait_asynccnt 0" ::: "memory");
        }

        // Consume tile s from buffer (s&1).
        const int      rl  = lane >> 4;                  // which of the 2 rows
        const int      h4  = (lane & 15) * 4;            // float4 group within H=64
        const unsigned row = r0 + 2u * (unsigned)s + (unsigned)rl;
        const int      b   = (int)(row >> 15);           // row / (NT*TL) = /32768
        const int      f   = foc[row];                   // 16 lanes share one load

        const float* src = wbase + (s & 1) * 128 + rl * 64 + h4;
        float4 v = *(const float4*)src;

        float* dst = out + ((size_t)(b * S_ + f)) * H_ + h4;
        unsafeAtomicAdd(dst + 0, v.x);
        unsafeAtomicAdd(dst + 1, v.y);
        unsafeAtomicAdd(dst + 2, v.z);
        unsafeAtomicAdd(dst + 3, v.w);
        if ((lane & 15) == 0)                            // one count bump per row
            atomicAdd(&cnt[b * S_ + f], 1);
    }
}

// ---------------------------------------------------------------------------
// Kernel C: divide by count, with count forced to 1 for pos >= seq_len[b].
// One thread per float4 of the output (1,048,576 threads).
// ---------------------------------------------------------------------------
__global__ __launch_bounds__(256) void divide_kernel(
    float* __restrict__ out, const int* __restrict__ cnt,
    const int* __restrict__ seq_lens)
{
    unsigned t  = blockIdx.x * 256u + threadIdx.x;   // < 1048576
    unsigned bs = t >> 4;                            // (b,s) id, < 65536
    unsigned h4 = (t & 15u) * 4u;
    unsigned b  = bs >> 12;                          // / S_ (4096)
    unsigned sp = bs & 4095u;

    int   c   = (sp >= (unsigned)seq_lens[b]) ? 1 : cnt[bs];
    float inv = 1.0f / (float)c;

    float4* p = (float4*)(out + (size_t)bs * H_ + h4);
    float4  v = *p;
    v.x *= inv; v.y *= inv; v.z *= inv; v.w *= inv;
    *p = v;
}

extern "C" void kernel_launch(void* const* d_in, const int* in_sizes, int n_in,
                              void* d_out, int out_size, void* d_ws, size_t ws_size,
                              hipStream_t stream) {
    const float* emb = (const float*)d_in[0];   // node_embeddings [B,NT,TL,H] f32
    const int*   foc = (const int*)  d_in[1];   // batched_focuses [B,NT,TL] i32
    const int*   seq = (const int*)  d_in[2];   // seq_lens [B] i32
    // d_in[3] = max_seq_len scalar; it is the compile-time constant S_.

    float* out = (float*)d_out;                 // [B,S,H] = 4,194,304 f32
    int*   cnt = (int*)d_ws;                    // [B,S]   = 65,536 i32 (256 KiB scratch)

    // A: zero output (1M uint4) and counts (16K uint4).
    zero4_kernel<<<4096, 256, 0, stream>>>((uint4*)out, 1048576u);
    zero4_kernel<<<64,   256, 0, stream>>>((uint4*)cnt, 16384u);

    // B: 524288 rows / 8 rows-per-wave / 8 waves-per-block = 8192 blocks.
    scatter_kernel<<<8192, 256, 0, stream>>>(emb, foc, out, cnt);

    // C: 4,194,304 floats / 4 per thread / 256 per block = 4096 blocks.
    divide_kernel<<<4096, 256, 0, stream>>>(out, cnt, seq);
}